// EdgeAttention_8615704396468
// MI455X (gfx1250) — compile-verified
//
#include <hip/hip_runtime.h>

typedef __attribute__((ext_vector_type(2))) float v2f;
typedef __attribute__((ext_vector_type(8))) float v8f;

#define N_NODES 4096
#define DIM     128

// Order-preserving float<->uint mapping so unsigned atomicMax gives float max.
__device__ __forceinline__ unsigned mapf(float f) {
    unsigned u = __float_as_uint(f);
    return (u & 0x80000000u) ? ~u : (u | 0x80000000u);
}
__device__ __forceinline__ float unmapf(unsigned u) {
    return __uint_as_float((u & 0x80000000u) ? (u ^ 0x80000000u) : ~u);
}

// -------- Kernel 1: T = F @ W^T via V_WMMA_F32_16X16X4_F32 --------
// One wave computes one 16x16 tile of T, accumulating K=128 in 32 WMMA steps.
// A-tile (16x4 f32): lane l holds row (l&15), K = kbase + 2*(l>>4) + {0,1}
// B-tile (4x16 f32): lane l holds col (l&15) = output feature (row of W),
//                    K = kbase + 2*(l>>4) + {0,1}  (same contiguous pair!)
__global__ void transform_wmma(const float* __restrict__ F,
                               const float* __restrict__ W,
                               float* __restrict__ T, int rows) {
    const int lane  = threadIdx.x & 31;
    const int wave  = threadIdx.x >> 5;
    const int tile  = blockIdx.x * (blockDim.x >> 5) + wave;   // wave-uniform
    const int etiles = DIM >> 4;                               // 8
    const int ntiles = (rows >> 4) * etiles;
    if (tile >= ntiles) return;                                // uniform exit: EXEC stays all-1s

    const int et  = tile % etiles;
    const int rt  = tile / etiles;
    const int l15 = lane & 15;
    const int g   = lane >> 4;

    const float* Arow = F + (size_t)((rt << 4) + l15) * DIM;   // feature row (M)
    const float* Brow = W + (size_t)((et << 4) + l15) * DIM;   // W row (output feature = N)

    v8f acc = {};
#pragma unroll
    for (int k = 0; k < DIM; k += 4) {
        v2f a, b;
        a.x = Arow[k + 2 * g];
        a.y = Arow[k + 2 * g + 1];
        b.x = Brow[k + 2 * g];
        b.y = Brow[k + 2 * g + 1];
        acc = __builtin_amdgcn_wmma_f32_16x16x4_f32(
            /*neg_a=*/false, a, /*neg_b=*/false, b,
            /*c_mod=*/(short)0, acc, /*reuse_a=*/false, /*reuse_b=*/false);
    }

    // D layout: VGPR v, lanes0-15 -> M=v, lanes16-31 -> M=v+8; N = lane&15
    float* Tcol = T + (size_t)(rt << 4) * DIM + (et << 4) + l15;
#pragma unroll
    for (int v = 0; v < 8; ++v)
        Tcol[(size_t)(v + 8 * g) * DIM] = acc[v];
}

// -------- Kernel 2: per-edge score (one wave per edge) + column max --------
__global__ void edge_scores(const float* __restrict__ F, const float* __restrict__ T,
                            const int* __restrict__ esrc, const int* __restrict__ edst,
                            float* __restrict__ escore, unsigned* __restrict__ colmax,
                            int Etot, int total) {
    const int lane = threadIdx.x & 31;
    const int idx  = blockIdx.x * (blockDim.x >> 5) + (threadIdx.x >> 5);
    if (idx >= total) return;
    const int b   = idx / Etot;
    const int src = esrc[idx];
    const int dst = edst[idx];
    const float4* fa = (const float4*)(F + ((size_t)b * N_NODES + src) * DIM);
    const float4* fb = (const float4*)(T + ((size_t)b * N_NODES + dst) * DIM);
    float4 x = fa[lane], y = fb[lane];
    float d = x.x * y.x + x.y * y.y + x.z * y.z + x.w * y.w;
#pragma unroll
    for (int off = 16; off > 0; off >>= 1)
        d += __shfl_xor(d, off, 32);
    if (lane == 0) {
        escore[idx] = d;
        atomicMax(colmax + b * N_NODES + dst, mapf(d));
    }
}

// -------- Kernel 3: deduplicated column sums of exp(s - max) --------
__global__ void edge_sum(const float* __restrict__ escore,
                         const int* __restrict__ esrc, const int* __restrict__ edst,
                         const unsigned* __restrict__ colmax, float* __restrict__ colsum,
                         unsigned* __restrict__ claim, int Etot, int total) {
    const int idx = blockIdx.x * blockDim.x + threadIdx.x;
    if (idx >= total) return;
    const int b   = idx / Etot;
    const int src = esrc[idx];
    const int dst = edst[idx];
    const size_t cell = ((size_t)b * N_NODES + src) * N_NODES + dst;
    const unsigned bit = 1u << (cell & 31);
    unsigned old = atomicOr(claim + (cell >> 5), bit);
    if (!(old & bit)) {   // first claimant of this (src,dst) cell
        const int col = b * N_NODES + dst;
        float m = unmapf(colmax[col]);
        atomicAdd(colsum + col, expf(escore[idx] - m));
    }
}

// -------- Kernel 4: scatter normalized values (duplicates write same value) --------
__global__ void edge_write(const float* __restrict__ escore,
                           const int* __restrict__ esrc, const int* __restrict__ edst,
                           const unsigned* __restrict__ colmax, const float* __restrict__ colsum,
                           float* __restrict__ out, int Etot, int total) {
    const int idx = blockIdx.x * blockDim.x + threadIdx.x;
    if (idx >= total) return;
    const int b   = idx / Etot;
    const int src = esrc[idx];
    const int dst = edst[idx];
    const int col = b * N_NODES + dst;
    const size_t cell = ((size_t)b * N_NODES + src) * N_NODES + dst;
    float m = unmapf(colmax[col]);
    out[cell] = expf(escore[idx] - m) / colsum[col];
}

extern "C" void kernel_launch(void* const* d_in, const int* in_sizes, int n_in,
                              void* d_out, int out_size, void* d_ws, size_t ws_size,
                              hipStream_t stream) {
    const float* F    = (const float*)d_in[0];
    const float* W    = (const float*)d_in[1];
    const int*   esrc = (const int*)d_in[2];
    const int*   edst = (const int*)d_in[3];
    float*       out  = (float*)d_out;

    const int rows   = in_sizes[0] / DIM;      // B*N = 16384
    const int B      = rows / N_NODES;         // 4
    const int totalE = in_sizes[2];            // B*(E+N) = 540672
    const int Etot   = totalE / B;             // 135168

    // Workspace layout
    char*  ws     = (char*)d_ws;
    float* T      = (float*)ws;
    size_t offE   = (size_t)rows * DIM * sizeof(float);            // 8 MB
    float* escore = (float*)(ws + offE);
    size_t offM   = offE + (size_t)totalE * sizeof(float);         // +2.06 MB
    unsigned* colmax = (unsigned*)(ws + offM);
    size_t offS   = offM + (size_t)rows * sizeof(unsigned);        // +64 KB
    float* colsum = (float*)(ws + offS);
    size_t offC   = offS + (size_t)rows * sizeof(float);           // +64 KB
    unsigned* claim = (unsigned*)(ws + offC);
    size_t claimWords = ((size_t)rows * N_NODES + 31) / 32;        // 8 MB

    // Output is zero except at edge cells; colmax=0 is "-inf" under mapf.
    hipMemsetAsync(d_out, 0, (size_t)out_size * sizeof(float), stream);
    hipMemsetAsync(ws + offM, 0,
                   (size_t)rows * 2 * sizeof(unsigned) + claimWords * sizeof(unsigned),
                   stream);

    const int tiles = (rows / 16) * (DIM / 16);                    // 8192 wave-tiles
    transform_wmma<<<(tiles + 7) / 8, 256, 0, stream>>>(F, W, T, rows);

    edge_scores<<<(totalE + 7) / 8, 256, 0, stream>>>(F, T, esrc, edst,
                                                      escore, colmax, Etot, totalE);

    edge_sum<<<(totalE + 255) / 256, 256, 0, stream>>>(escore, esrc, edst,
                                                       colmax, colsum, claim, Etot, totalE);

    edge_write<<<(totalE + 255) / 256, 256, 0, stream>>>(escore, esrc, edst,
                                                         colmax, colsum, out, Etot, totalE);
}